// PointPillars_85194971283674
// MI455X (gfx1250) — compile-verified
//
#include <hip/hip_runtime.h>

#define Hh 248
#define Ww 216
#define HW 53568                 // 248*216
#define NA (HW*6)                // 321408 anchors per batch
#define BSZ 16
#define NPRE 100
#define MAXN 50
#define PI_F 3.14159265358979323846f

typedef unsigned int uint;
typedef unsigned long long u64;
typedef unsigned int v4u __attribute__((ext_vector_type(4)));
typedef int          v8i __attribute__((ext_vector_type(8)));
typedef int          v4i __attribute__((ext_vector_type(4)));

// monotone map float -> uint (order preserving over full float range)
__device__ __forceinline__ uint mono_key(float f) {
  uint u = __float_as_uint(f);
  return (u & 0x80000000u) ? ~u : (u | 0x80000000u);
}

// ---------------------------------------------------------------------------
// Kernel 1: stream cls_pred via TDM tiles -> per-anchor max logit into ws
// ---------------------------------------------------------------------------
#define K1_TILE 512

__global__ __launch_bounds__(K1_TILE) void k1_scores(const float* __restrict__ cls,
                                                     float* __restrict__ scores) {
  __shared__ float tile[18 * K1_TILE];            // 36 KB
  const int chunk = blockIdx.x;
  const int b     = blockIdx.y;
  const int s0    = chunk * K1_TILE;
  const int remain = HW - s0;                     // >0 by grid construction

#if __has_builtin(__builtin_amdgcn_tensor_load_to_lds)
  if (threadIdx.x == 0) {                         // one TDM issue per workgroup (wave 0)
    unsigned long long gaddr =
        (unsigned long long)(const char*)(cls + (size_t)b * 18 * HW + s0);
    uint lds = (uint)(unsigned long long)(&tile[0]);   // flat LDS aperture: low 32 bits = LDS offset
    // D# group0: count=1, lds_addr, global_addr[56:0], type=2
    v4u g0;
    g0.x = 1u;
    g0.y = lds;
    g0.z = (uint)gaddr;
    g0.w = ((uint)(gaddr >> 32) & 0x1FFFFFFu) | 0x80000000u;
    // D# group1: data_size=4B; tensor_dim0=remain; tensor_dim1=18;
    //            tile_dim0=512; tile_dim1=18; tensor_dim0_stride=HW
    uint dim0 = (uint)remain;
    v8i g1;
    g1[0] = (int)(2u << 16);                       // data_size=2 -> 4 bytes
    g1[1] = (int)((dim0 & 0xFFFFu) << 16);         // tensor_dim0[15:0] at bits 63:48
    g1[2] = (int)((dim0 >> 16) | (18u << 16));     // tensor_dim0[31:16] | tensor_dim1 lo
    g1[3] = (int)((uint)K1_TILE << 16);            // tensor_dim1 hi=0 | tile_dim0
    g1[4] = 18;                                    // tile_dim1=18, tile_dim2=0
    g1[5] = (int)(uint)HW;                         // tensor_dim0_stride lo32
    g1[6] = 0;                                     // stride hi / dim1_stride lo
    g1[7] = 0;
    v4i gz = {0, 0, 0, 0};
#if defined(__clang_major__) && (__clang_major__ >= 23)
    v8i gz8 = {0, 0, 0, 0, 0, 0, 0, 0};
    __builtin_amdgcn_tensor_load_to_lds(g0, g1, gz, gz, gz8, 0);
#else
    __builtin_amdgcn_tensor_load_to_lds(g0, g1, gz, gz, 0);
#endif
    __builtin_amdgcn_s_wait_tensorcnt(0);
  }
#else
  // fallback: cooperative LDS fill
  for (int idx = threadIdx.x; idx < 18 * K1_TILE; idx += K1_TILE) {
    int c = idx / K1_TILE, x = idx - c * K1_TILE;
    tile[idx] = (x < remain) ? cls[(size_t)b * 18 * HW + (size_t)c * HW + s0 + x] : 0.0f;
  }
#endif
  __syncthreads();

  const int sl = threadIdx.x;
  const int s  = s0 + sl;
  if (s < HW) {
    float* out = scores + (size_t)b * NA + (size_t)s * 6;
#pragma unroll
    for (int a = 0; a < 6; ++a) {
      float v0 = tile[(a * 3 + 0) * K1_TILE + sl];
      float v1 = tile[(a * 3 + 1) * K1_TILE + sl];
      float v2 = tile[(a * 3 + 2) * K1_TILE + sl];
      out[a] = fmaxf(v0, fmaxf(v1, v2));
    }
  }
}

// ---------------------------------------------------------------------------
// Kernel 2: exact deterministic top-100 per batch (histogram threshold select)
// ---------------------------------------------------------------------------
#define K2_T 1024

__global__ __launch_bounds__(K2_T) void k2_select(const float* __restrict__ scores,
                                                  int* __restrict__ sel) {
  __shared__ uint hist[4096];     // 12-bit monotone-key bins
  __shared__ uint sA[K2_T];
  __shared__ uint sB[K2_T];
  __shared__ uint cand_key[1024];
  __shared__ uint cand_idx[1024];
  __shared__ int  shB, shNa;

  const int t = threadIdx.x;
  const int b = blockIdx.x;
  const float* sc = scores + (size_t)b * NA;
  int* selb = sel + b * NPRE;

  for (int i = t; i < 4096; i += K2_T) hist[i] = 0u;
  __syncthreads();
  for (int i = t; i < NA; i += K2_T)
    atomicAdd(&hist[mono_key(sc[i]) >> 20], 1u);
  __syncthreads();

  // suffix sums over 1024 segments of 4 bins each
  sA[t] = hist[4 * t] + hist[4 * t + 1] + hist[4 * t + 2] + hist[4 * t + 3];
  __syncthreads();
  for (int off = 1; off < K2_T; off <<= 1) {
    uint v = (t + off < K2_T) ? sA[t + off] : 0u;
    __syncthreads();
    sA[t] += v;
    __syncthreads();
  }
  uint Sself = sA[t];
  uint Snext = (t < K2_T - 1) ? sA[t + 1] : 0u;
  if (Sself >= NPRE && Snext < NPRE) {          // unique crossing thread
    uint c = Snext;
    for (int bn = 4 * t + 3; bn >= 4 * t; --bn) {
      c += hist[bn];
      if (c >= NPRE) { shB = bn; shNa = (int)(c - hist[bn]); break; }
    }
  }
  __syncthreads();
  const uint B  = (uint)shB;
  const int  Na = shNa;                         // strictly-above count (< 100)
  const int  k  = NPRE - Na;                    // needed from threshold bucket

  // count per thread: definite (bin>B) and candidates (bin==B)
  uint cd = 0, cc2 = 0;
  for (int i = t; i < NA; i += K2_T) {
    uint bn = mono_key(sc[i]) >> 20;
    cd  += (bn > B);
    cc2 += (bn == B);
  }
  sA[t] = cd; sB[t] = cc2;
  __syncthreads();
  for (int off = 1; off < K2_T; off <<= 1) {    // inclusive prefix scans
    uint vd = (t >= off) ? sA[t - off] : 0u;
    uint vc = (t >= off) ? sB[t - off] : 0u;
    __syncthreads();
    sA[t] += vd; sB[t] += vc;
    __syncthreads();
  }
  uint dpos = sA[t] - cd;
  uint cpos = sB[t] - cc2;
  for (int i = t; i < NA; i += K2_T) {          // deterministic compaction
    uint kk = mono_key(sc[i]);
    uint bn = kk >> 20;
    if (bn > B) {
      selb[dpos++] = i;
    } else if (bn == B) {
      if (cpos < 1024u) { cand_key[cpos] = kk; cand_idx[cpos] = (uint)i; }
      cpos++;
    }
  }
  __syncthreads();

  uint C = sB[K2_T - 1];
  if (C > 1024u) C = 1024u;
  if (t < 32) {                                 // wave-0 register tournament
    u64 loc[32];
    int nloc = 0;
    for (uint s2 = (uint)t; s2 < C; s2 += 32u)
      loc[nloc++] = ((u64)cand_key[s2] << 32) | (u64)(0xFFFFFFFFu - cand_idx[s2]);
    for (int j = 0; j < k; ++j) {
      u64 best = 0; int bq = -1;
      for (int q = 0; q < nloc; ++q)
        if (loc[q] > best) { best = loc[q]; bq = q; }
      u64 mine = best;
      for (int o = 16; o > 0; o >>= 1) {
        u64 other = __shfl_xor(best, o, 32);
        if (other > best) best = other;
      }
      if (bq >= 0 && mine == best) loc[bq] = 0; // unique winner invalidates
      if (t == 0) {
        int idx = best ? (int)(0xFFFFFFFFu - (uint)(best & 0xFFFFFFFFull)) : 0;
        selb[Na + j] = idx;
      }
    }
  }
}

// ---------------------------------------------------------------------------
// Kernel 3: decode + per-class greedy NMS + sorted top-50 outputs
// ---------------------------------------------------------------------------
__global__ __launch_bounds__(128) void k3_nms(const float* __restrict__ cls,
                                              const float* __restrict__ box,
                                              const float* __restrict__ dir,
                                              const int* __restrict__ sel,
                                              float* __restrict__ out) {
  __shared__ float sc3[3 * NPRE];
  __shared__ float b2d[NPRE * 4];
  __shared__ float area[NPRE];
  __shared__ float bb7[NPRE * 7];
  __shared__ int   kept[3 * NPRE];
  __shared__ int   processed[NPRE];
  __shared__ float flat[3 * NPRE];
  __shared__ int   sh_pivot;

  const int t = threadIdx.x;
  const int b = blockIdx.x;

  if (t < NPRE) {
    int i  = sel[b * NPRE + t];
    int s  = i / 6, a = i - s * 6;
    int c  = a >> 1, r = a & 1;
    int iy = s / Ww, ix = s - iy * Ww;
    const float z0s[3] = {-0.6f, -0.6f, -1.78f};
    const float szw[3] = {0.6f, 0.6f, 1.6f};
    const float szl[3] = {0.8f, 1.76f, 3.9f};
    const float szh[3] = {1.73f, 1.73f, 1.56f};
    float xa = ((float)ix + 0.5f) * (69.12f / 216.0f);
    float ya = ((float)iy + 0.5f) * (79.36f / 248.0f) - 39.68f;
    float za = z0s[c], wa = szw[c], la = szl[c], ha = szh[c];
    float ra = r ? 1.57f : 0.0f;
    const float* boxb = box + (size_t)b * 42 * HW;
    const float* dirb = dir + (size_t)b * 12 * HW;
    const float* clsb = cls + (size_t)b * 18 * HW;
    float dx  = boxb[(a * 7 + 0) * HW + s];
    float dy  = boxb[(a * 7 + 1) * HW + s];
    float dz  = boxb[(a * 7 + 2) * HW + s];
    float dw  = boxb[(a * 7 + 3) * HW + s];
    float dl  = boxb[(a * 7 + 4) * HW + s];
    float dh  = boxb[(a * 7 + 5) * HW + s];
    float drv = boxb[(a * 7 + 6) * HW + s];
    float da = sqrtf(wa * wa + la * la);
    float x  = dx * da + xa;
    float y  = dy * da + ya;
    float z  = dz * ha + za + ha * 0.5f;
    float w  = wa * expf(dw);
    float l  = la * expf(dl);
    float h  = ha * expf(dh);
    z -= h * 0.5f;
    float t0 = ra + drv;
    float d0 = dirb[(a * 2 + 0) * HW + s];
    float d1 = dirb[(a * 2 + 1) * HW + s];
    int dircls = (d1 > d0) ? 1 : 0;
    float lp = t0 - floorf(t0 / PI_F + 1.0f) * PI_F;
    float theta = lp + (1.0f - (float)dircls) * PI_F;
    bb7[t * 7 + 0] = x;  bb7[t * 7 + 1] = y;  bb7[t * 7 + 2] = z;
    bb7[t * 7 + 3] = w;  bb7[t * 7 + 4] = l;  bb7[t * 7 + 5] = h;
    bb7[t * 7 + 6] = theta;
    b2d[t * 4 + 0] = x - w * 0.5f;  b2d[t * 4 + 1] = y - l * 0.5f;
    b2d[t * 4 + 2] = x + w * 0.5f;  b2d[t * 4 + 3] = y + l * 0.5f;
    area[t] = w * l;
#pragma unroll
    for (int kcl = 0; kcl < 3; ++kcl)
      sc3[kcl * NPRE + t] = 1.0f / (1.0f + expf(-clsb[(a * 3 + kcl) * HW + s]));
  }
  __syncthreads();

  // per-class greedy NMS, equivalent to sorted-order suppression
  for (int ccl = 0; ccl < 3; ++ccl) {
    if (t < NPRE) {
      kept[ccl * NPRE + t] = sc3[ccl * NPRE + t] > 0.1f;
      processed[t] = 0;
    }
    __syncthreads();
    for (int it = 0; it < NPRE; ++it) {
      if (t < 32) {
        u64 best = 0;
        for (int j = t; j < NPRE; j += 32) {
          if (kept[ccl * NPRE + j] && !processed[j]) {
            u64 kk = ((u64)__float_as_uint(sc3[ccl * NPRE + j]) << 32) |
                     (u64)(NPRE - j);               // sigmoid > 0 -> bits monotone
            if (kk > best) best = kk;
          }
        }
        for (int o = 16; o > 0; o >>= 1) {
          u64 other = __shfl_xor(best, o, 32);
          if (other > best) best = other;
        }
        if (t == 0)
          sh_pivot = best ? (NPRE - (int)(uint)(best & 0xFFFFFFFFull)) : -1;
      }
      __syncthreads();
      int p = sh_pivot;
      if (p < 0) break;                             // uniform
      if (t == 0) processed[p] = 1;
      if (t < NPRE && t != p && kept[ccl * NPRE + t] && !processed[t]) {
        float x1 = fmaxf(b2d[p * 4 + 0], b2d[t * 4 + 0]);
        float y1 = fmaxf(b2d[p * 4 + 1], b2d[t * 4 + 1]);
        float x2 = fminf(b2d[p * 4 + 2], b2d[t * 4 + 2]);
        float y2 = fminf(b2d[p * 4 + 3], b2d[t * 4 + 3]);
        float inter = fmaxf(x2 - x1, 0.0f) * fmaxf(y2 - y1, 0.0f);
        float iou = inter / (area[p] + area[t] - inter + 1e-8f);
        if (iou > 0.01f) kept[ccl * NPRE + t] = 0;
      }
      __syncthreads();
    }
    __syncthreads();
  }

  for (int j = t; j < 3 * NPRE; j += 128)
    flat[j] = kept[j] ? sc3[j] : -1.0f;
  __syncthreads();

  if (t < 32) {                                     // sorted top-50 picks
    volatile float* vf = flat;
    for (int rr = 0; rr < MAXN; ++rr) {
      u64 best = 0;
      for (int j = t; j < 3 * NPRE; j += 32) {
        u64 kk = ((u64)mono_key(vf[j]) << 32) | (u64)(4096 - j);
        if (kk > best) best = kk;
      }
      for (int o = 16; o > 0; o >>= 1) {
        u64 other = __shfl_xor(best, o, 32);
        if (other > best) best = other;
      }
      if (t == 0) {
        int fi = 4096 - (int)(uint)(best & 0xFFFFFFFFull);
        float val = vf[fi];
        int lab = fi / NPRE;
        int sj  = fi - lab * NPRE;
        float* ob = out + ((size_t)b * MAXN + rr) * 7;
        for (int q = 0; q < 7; ++q) ob[q] = bb7[sj * 7 + q];
        out[BSZ * MAXN * 7 + b * MAXN + rr]                  = (float)lab;
        out[BSZ * MAXN * 7 + BSZ * MAXN + b * MAXN + rr]     = val;
        out[BSZ * MAXN * 7 + 2 * BSZ * MAXN + b * MAXN + rr] = (val > 0.0f) ? 1.0f : 0.0f;
        vf[fi] = -__builtin_inff();
      }
    }
  }
}

// ---------------------------------------------------------------------------
extern "C" void kernel_launch(void* const* d_in, const int* in_sizes, int n_in,
                              void* d_out, int out_size, void* d_ws, size_t ws_size,
                              hipStream_t stream) {
  const float* cls = (const float*)d_in[0];
  const float* box = (const float*)d_in[1];
  const float* dir = (const float*)d_in[2];
  // d_in[3] (anchors) recomputed analytically on device.

  float* scores = (float*)d_ws;                              // 16*NA floats (~20.6 MB)
  int*   sel    = (int*)((char*)d_ws + (size_t)BSZ * NA * sizeof(float)); // 16*100 ints

  dim3 g1((HW + K1_TILE - 1) / K1_TILE, BSZ);
  k1_scores<<<g1, K1_TILE, 0, stream>>>(cls, scores);
  k2_select<<<BSZ, K2_T, 0, stream>>>(scores, sel);
  k3_nms<<<BSZ, 128, 0, stream>>>(cls, box, dir, sel, (float*)d_out);
}